// Network_50122268345056
// MI455X (gfx1250) — compile-verified
//
#include <hip/hip_runtime.h>
#include <math.h>

typedef __attribute__((ext_vector_type(2))) float v2f;
typedef __attribute__((ext_vector_type(8))) float v8f;

#define INV_NN     0.512989176042577f    /* 1/sqrt(3.8) */
#define INV_SQRT3  0.5773502691896258f
#define S3   1.7320508075688772f
#define S5   2.23606797749979f
#define S7   2.6457513110645907f
#define S15  3.872983346207417f
#define S105 10.246950765959598f
#define C358 2.091650066335189f          /* sqrt(35/8) */
#define C218 1.6201851746019651f         /* sqrt(21/8) */

__global__ void kzero(float* __restrict__ p, long n) {
    long i = (long)blockIdx.x * blockDim.x + threadIdx.x;
    if (i < n) p[i] = 0.0f;
}

// ---------------- Kernel A: per-edge geometry, scatter x[N,16] ----------------
__global__ void kA(const float* __restrict__ evec, const float* __restrict__ elen,
                   const int* __restrict__ edst,
                   float* __restrict__ sh, float* __restrict__ basis,
                   float* __restrict__ x, int E)
{
    int e = blockIdx.x * blockDim.x + threadIdx.x;
    if (e >= E) return;
    float vx = evec[3*(size_t)e], vy = evec[3*(size_t)e+1], vz = evec[3*(size_t)e+2];
    float rn = 1.0f / (sqrtf(vx*vx + vy*vy + vz*vz) + 1e-12f);
    vx *= rn; vy *= rn; vz *= rn;
    float x2 = vx*vx, y2 = vy*vy, z2 = vz*vz;
    float s[16];
    s[0]  = 1.0f;
    s[1]  = S3*vx; s[2] = S3*vy; s[3] = S3*vz;
    s[4]  = S15*vx*vz;
    s[5]  = S15*vx*vy;
    s[6]  = 0.5f*S5*(2.0f*y2 - x2 - z2);
    s[7]  = S15*vy*vz;
    s[8]  = 0.5f*S15*(z2 - x2);
    s[9]  = C358*vx*(3.0f*z2 - x2);
    s[10] = S105*vx*vy*vz;
    s[11] = C218*vx*(4.0f*y2 - z2 - x2);
    s[12] = 0.5f*S7*vy*(2.0f*y2 - 3.0f*z2 - 3.0f*x2);
    s[13] = C218*vz*(4.0f*y2 - z2 - x2);
    s[14] = 0.5f*S105*vy*(z2 - x2);
    s[15] = C358*vz*(z2 - 3.0f*x2);
    int d = edst[e];
    #pragma unroll
    for (int j = 0; j < 16; j++) {
        sh[(size_t)e*16 + j] = s[j];
        atomicAdd(&x[(size_t)d*16 + j], s[j] * INV_NN);
    }
    float len = elen[e];
    #pragma unroll
    for (int i = 0; i < 3; i++) {
        float dd = (len - 2.0f*(float)i) * 0.5f;
        basis[(size_t)e*3 + i] = expf(-dd*dd) * (1.0f/1.12f);
    }
}

// ---- Kernel B: per-edge MLP1 GEMM [32e x 256] x [256 x 384] via f32 WMMA ----
// Block = 192 threads (6 waves); block owns TWO 16-edge sub-tiles (32 edges);
// wave w owns output columns [16w,16w+16) of ef[*,96]. Each B fragment loaded
// from W1b is reused by 2 WMMAs (one per edge sub-tile): 8 loads : 8 wmma per
// K-step, and W1b L2 traffic is halved vs one tile per block.
__global__ __launch_bounds__(192) void kB(
    const float* __restrict__ sh, const float* __restrict__ basis,
    const float* __restrict__ x,
    const int* __restrict__ esrc, const int* __restrict__ edst,
    const float* __restrict__ W1a, const float* __restrict__ W1b,
    float* __restrict__ h, int E)
{
    __shared__ float h1[32][258];   // padded: bank = (258*m + c) % 64 varies with m
    __shared__ float dots[32][4];
    __shared__ int   dstT[32];

    const int tid = threadIdx.x;
    const int e0  = blockIdx.x * 32;

    // Cooperative hidden layer: h1[m][c] = relu((basis[e]·W1a[:,c]) / sqrt(3))
    for (int idx = tid; idx < 32*256; idx += 192) {
        int m = idx >> 8, c = idx & 255;
        int e = e0 + m;
        float v = 0.0f;
        if (e < E) {
            float b0 = basis[(size_t)e*3], b1 = basis[(size_t)e*3+1], b2 = basis[(size_t)e*3+2];
            v = b0*W1a[c] + b1*W1a[256+c] + b2*W1a[512+c];
            v = fmaxf(v * INV_SQRT3, 0.0f);
        }
        h1[m][c] = v;
    }
    // dots[m][p] = (x[src] . sh[e]) over slice p, / sqrt(2p+1)
    if (tid < 128) {
        int m = tid >> 2, p = tid & 3;
        int e = e0 + m;
        float s = 0.0f;
        if (e < E) {
            int src = esrc[e];
            int a = p*p, b = a + 2*p + 1;
            for (int j = a; j < b; j++)
                s += x[(size_t)src*16 + j] * sh[(size_t)e*16 + j];
            s *= rsqrtf((float)(2*p + 1));
        }
        dots[m][p] = s;
        if (p == 0) dstT[m] = (e < E) ? edst[e] : 0;
    }
    __syncthreads();

    const int kc    = tid >> 5;        // wave id 0..5 = output column chunk
    const int lane  = tid & 31;
    const int mA    = lane & 15;       // A-matrix row (edge within sub-tile)
    const int khalf = lane >> 4;       // K sub-pair select
    const int n     = lane & 15;       // B/D column within chunk

    v8f a00 = {}, a01 = {}, a02 = {}, a03 = {};  // sub-tile 0, p = 0..3
    v8f a10 = {}, a11 = {}, a12 = {}, a13 = {};  // sub-tile 1, p = 0..3
    const int colb = kc * 16 + n;

    for (int k0 = 0; k0 < 256; k0 += 4) {
        int kb = k0 + 2*khalf;
        v2f A0, A1;
        A0.x = h1[mA][kb];      A0.y = h1[mA][kb + 1];
        A1.x = h1[16 + mA][kb]; A1.y = h1[16 + mA][kb + 1];
        const float* Brow0 = W1b + (size_t)kb * 384;
        const float* Brow1 = Brow0 + 384;
        v2f B0; B0.x = Brow0[colb];       B0.y = Brow1[colb];
        v2f B1; B1.x = Brow0[96  + colb]; B1.y = Brow1[96  + colb];
        v2f B2; B2.x = Brow0[192 + colb]; B2.y = Brow1[192 + colb];
        v2f B3; B3.x = Brow0[288 + colb]; B3.y = Brow1[288 + colb];
        a00 = __builtin_amdgcn_wmma_f32_16x16x4_f32(false, A0, false, B0, (short)0, a00, false, false);
        a10 = __builtin_amdgcn_wmma_f32_16x16x4_f32(false, A1, false, B0, (short)0, a10, false, false);
        a01 = __builtin_amdgcn_wmma_f32_16x16x4_f32(false, A0, false, B1, (short)0, a01, false, false);
        a11 = __builtin_amdgcn_wmma_f32_16x16x4_f32(false, A1, false, B1, (short)0, a11, false, false);
        a02 = __builtin_amdgcn_wmma_f32_16x16x4_f32(false, A0, false, B2, (short)0, a02, false, false);
        a12 = __builtin_amdgcn_wmma_f32_16x16x4_f32(false, A1, false, B2, (short)0, a12, false, false);
        a03 = __builtin_amdgcn_wmma_f32_16x16x4_f32(false, A0, false, B3, (short)0, a03, false, false);
        a13 = __builtin_amdgcn_wmma_f32_16x16x4_f32(false, A1, false, B3, (short)0, a13, false, false);
    }

    // ef[m,n] = sum_p dots[m][p]*acc_p ; scale = (1/16 from W1b norm)*0.5*inv_nn
    const float scale = (1.0f/16.0f) * 0.5f * INV_NN;
    #pragma unroll
    for (int r = 0; r < 8; r++) {
        int m = r + 8*khalf;            // D layout: VGPR r holds M=r / M=r+8
        if (e0 + m < E) {
            float ef = dots[m][0]*a00[r] + dots[m][1]*a01[r]
                     + dots[m][2]*a02[r] + dots[m][3]*a03[r];
            atomicAdd(&h[(size_t)dstT[m]*96 + kc*16 + n], ef * scale);
        }
        int m1 = 16 + m;
        if (e0 + m1 < E) {
            float ef = dots[m1][0]*a10[r] + dots[m1][1]*a11[r]
                     + dots[m1][2]*a12[r] + dots[m1][3]*a13[r];
            atomicAdd(&h[(size_t)dstT[m1]*96 + kc*16 + n], ef * scale);
        }
    }
}

// ---------------- Kernel C: node gated nonlinearity -> y[N,64] ----------------
__global__ void kC(const float* __restrict__ h, float* __restrict__ y, int N)
{
    int nid = blockIdx.x * blockDim.x + threadIdx.x;
    if (nid >= N) return;
    const float* hr = h + (size_t)nid * 96;
    float* yr = y + (size_t)nid * 64;
    #pragma unroll
    for (int j = 0; j < 16; j++) yr[j] = fmaxf(hr[j], 0.0f);
    #pragma unroll
    for (int j = 16; j < 32; j++) yr[j] = fabsf(hr[j]);
    #pragma unroll
    for (int j = 0; j < 32; j++) {
        float g = hr[32 + j];
        bool use_relu = (j < 8) || (j >= 16 && j < 24);
        float ga = use_relu ? fmaxf(g, 0.0f) : tanhf(g);
        yr[32 + j] = ga * hr[64 + j];
    }
}

// ---------------- Kernel D: z[N,256] = y[N,64] @ W2b^T  (W2b is [256,64]) ----
__global__ __launch_bounds__(256) void kD(const float* __restrict__ y,
                                          const float* __restrict__ W2b,
                                          float* __restrict__ z, int N)
{
    __shared__ float yn[64];
    int nid = blockIdx.x;
    int c = threadIdx.x;
    if (c < 64) yn[c] = y[(size_t)nid*64 + c];
    __syncthreads();
    float s = 0.0f;
    const float* wr = W2b + (size_t)c * 64;
    #pragma unroll 8
    for (int k = 0; k < 64; k++) s += wr[k] * yn[k];
    z[(size_t)nid*256 + c] = s;
}

// ---- Kernel E: per-edge h2 = relu(basis@W2a/sqrt3); out[dst] += h2.z[src] ----
__global__ __launch_bounds__(256) void kE(const float* __restrict__ basis,
                                          const float* __restrict__ z,
                                          const int* __restrict__ esrc,
                                          const int* __restrict__ edst,
                                          const float* __restrict__ W2a,
                                          float* __restrict__ out, int E)
{
    int e = blockIdx.x * (blockDim.x >> 5) + (threadIdx.x >> 5);
    int lane = threadIdx.x & 31;
    if (e >= E) return;
    float b0 = basis[(size_t)e*3], b1 = basis[(size_t)e*3+1], b2 = basis[(size_t)e*3+2];
    int src = esrc[e];
    const float* zr = z + (size_t)src * 256;
    float s = 0.0f;
    #pragma unroll
    for (int i = 0; i < 8; i++) {
        int c = lane + 32*i;
        float hv = b0*W2a[c] + b1*W2a[256+c] + b2*W2a[512+c];
        hv = fmaxf(hv * INV_SQRT3, 0.0f);
        s += hv * zr[c];
    }
    #pragma unroll
    for (int off = 16; off > 0; off >>= 1) s += __shfl_down(s, off, 32);
    if (lane == 0)
        atomicAdd(&out[edst[e]], s * (1.0f/128.0f) * INV_NN); // 1/16 (W2b) * 1/8
}

extern "C" void kernel_launch(void* const* d_in, const int* in_sizes, int n_in,
                              void* d_out, int out_size, void* d_ws, size_t ws_size,
                              hipStream_t stream)
{
    const float* evec = (const float*)d_in[0];
    const float* elen = (const float*)d_in[1];
    const float* W1a  = (const float*)d_in[2];
    const float* W1b  = (const float*)d_in[3];
    const float* W2a  = (const float*)d_in[4];
    const float* W2b  = (const float*)d_in[5];
    const int*   esrc = (const int*)d_in[6];
    const int*   edst = (const int*)d_in[7];
    float* out = (float*)d_out;

    const int E = in_sizes[0] / 3;
    const int N = out_size;

    float* ws = (float*)d_ws;
    float* x_     = ws;                 ws += (size_t)N * 16;
    float* h_     = ws;                 ws += (size_t)N * 96;
    float* y_     = ws;                 ws += (size_t)N * 64;
    float* z_     = ws;                 ws += (size_t)N * 256;
    float* sh_    = ws;                 ws += (size_t)E * 16;
    float* basis_ = ws;                 ws += (size_t)E * 3;

    long nx = (long)N * 16, nh = (long)N * 96;
    kzero<<<(int)((nx + 255) / 256), 256, 0, stream>>>(x_, nx);
    kzero<<<(int)((nh + 255) / 256), 256, 0, stream>>>(h_, nh);
    kzero<<<(N + 255) / 256, 256, 0, stream>>>(out, N);

    kA<<<(E + 255) / 256, 256, 0, stream>>>(evec, elen, edst, sh_, basis_, x_, E);
    kB<<<(E + 31) / 32, 192, 0, stream>>>(sh_, basis_, x_, esrc, edst, W1a, W1b, h_, E);
    kC<<<(N + 255) / 256, 256, 0, stream>>>(h_, y_, N);
    kD<<<N, 256, 0, stream>>>(y_, W2b, z_, N);
    kE<<<(E + 7) / 8, 256, 0, stream>>>(basis_, z_, esrc, edst, W2a, out, E);
}